// GeomLayer_MC_12506944766368
// MI455X (gfx1250) — compile-verified
//
#include <hip/hip_runtime.h>

// Problem constants (from reference): B=8, IN_CHAN=64, CHAN=64, H=W=56
#define NB      8
#define NIC     64
#define NC      64
#define HH      56
#define WW      56
#define HWSZ    (HH * WW)                // 3136 pixels
#define WG      256                      // 8 waves of 32
#define NCHUNK  4                        // 4 chunks of 16 input channels
#define CPLANES 16                       // planes staged per chunk
#define CHUNK_V4 (CPLANES * HWSZ / 4)    // 12544 float4 per chunk
#define V4_PER_THR (CHUNK_V4 / WG)       // 49 async b128 per thread per chunk
#define NTILES  (HWSZ / 16)              // 196 pixel tiles of 16 (exact)
#define TPW     25                       // max tiles per wave (ceil(196/8))

// Reciprocal constants: avoid IEEE fdiv chains in the hot loop.
#define INV_W2  (2.0f / (float)WW)       // 2/56
#define INV_H2  (2.0f / (float)HH)

typedef float v2f __attribute__((ext_vector_type(2)));
typedef float v8f __attribute__((ext_vector_type(8)));

// ---- CDNA5 async global->LDS copy (ASYNCcnt path) -------------------------
__device__ __forceinline__ void async_ld_b128(unsigned lds_off, const void* gaddr) {
    asm volatile("global_load_async_to_lds_b128 %0, %1, off"
                 :: "v"(lds_off), "v"(gaddr) : "memory");
}

#if __has_builtin(__builtin_amdgcn_s_wait_asynccnt)
#define WAIT_ASYNC(n) __builtin_amdgcn_s_wait_asynccnt(n)
#else
#define WAIT_ASYNC(n) asm volatile("s_wait_asynccnt %0" :: "i"(n) : "memory")
#endif

// Bilinear gather from one LDS-resident 56x56 plane; fx/fy are pixel-space
// coordinates. Matches reference semantics: weights/validity from unclipped
// corner positions, indices clipped.
__device__ __forceinline__ float bilin_gather(const float* __restrict__ buf,
                                              float fx, float fy) {
    float fx0 = floorf(fx), fy0 = floorf(fy);
    float ax = fx - fx0,    ay = fy - fy0;
    float wx0 = 1.0f - ax,  wx1 = ax;
    float wy0 = 1.0f - ay,  wy1 = ay;
    bool vx0 = (fx0 >=  0.0f) && (fx0 <= (float)(WW - 1));
    bool vx1 = (fx0 >= -1.0f) && (fx0 <= (float)(WW - 2));
    bool vy0 = (fy0 >=  0.0f) && (fy0 <= (float)(HH - 1));
    bool vy1 = (fy0 >= -1.0f) && (fy0 <= (float)(HH - 2));
    int qx0 = (int)fminf(fmaxf(fx0,        0.0f), (float)(WW - 1));
    int qx1 = (int)fminf(fmaxf(fx0 + 1.0f, 0.0f), (float)(WW - 1));
    int qy0 = (int)fminf(fmaxf(fy0,        0.0f), (float)(HH - 1));
    int qy1 = (int)fminf(fmaxf(fy0 + 1.0f, 0.0f), (float)(HH - 1));
    float v00 = buf[qy0 * WW + qx0];
    float v01 = buf[qy1 * WW + qx0];
    float v10 = buf[qy0 * WW + qx1];
    float v11 = buf[qy1 * WW + qx1];
    float s = 0.0f;
    s = fmaf((vx0 && vy0) ? wx0 * wy0 : 0.0f, v00, s);
    s = fmaf((vx0 && vy1) ? wx0 * wy1 : 0.0f, v01, s);
    s = fmaf((vx1 && vy0) ? wx1 * wy0 : 0.0f, v10, s);
    s = fmaf((vx1 && vy1) ? wx1 * wy1 : 0.0f, v11, s);
    return s;
}

// Sum of valid bilinear corner weights (the "box" term, gather-free).
__device__ __forceinline__ float box_weight(float fx, float fy) {
    float fx0 = floorf(fx), fy0 = floorf(fy);
    float ax = fx - fx0,    ay = fy - fy0;
    float wx0 = 1.0f - ax,  wx1 = ax;
    float wy0 = 1.0f - ay,  wy1 = ay;
    bool vx0 = (fx0 >=  0.0f) && (fx0 <= (float)(WW - 1));
    bool vx1 = (fx0 >= -1.0f) && (fx0 <= (float)(WW - 2));
    bool vy0 = (fy0 >=  0.0f) && (fy0 <= (float)(HH - 1));
    bool vy1 = (fy0 >= -1.0f) && (fy0 <= (float)(HH - 2));
    float bx = 0.0f;
    bx += (vx0 && vy0) ? wx0 * wy0 : 0.0f;
    bx += (vx0 && vy1) ? wx0 * wy1 : 0.0f;
    bx += (vx1 && vy0) ? wx1 * wy0 : 0.0f;
    bx += (vx1 && vy1) ? wx1 * wy1 : 0.0f;
    return bx;
}

// ---------------------------------------------------------------------------
__global__ __launch_bounds__(WG) void geom_layer_kernel(
    const float* __restrict__ x,      // (B, IC, H, W)
    const float* __restrict__ geo,    // (C, IC, 2, 3)
    const float* __restrict__ lin,    // (C, IC)
    const float* __restrict__ boxp,   // (C, 2, 3)
    float* __restrict__ out)          // (B, C, H, W)
{
    extern __shared__ float smem[];   // 16 planes * 3136 floats = 196 KB

    const int tid    = threadIdx.x;
    const int lane   = tid & 31;
    const int wave   = tid >> 5;
    const int laneLo = lane & 15;           // pixel slot within 16-pixel tile
    const bool hiHalf = (lane >= 16);       // handles channel rows 8..15 of chunk

    const int blk = blockIdx.x;             // grid = B*C, b-major for L2 reuse
    const int b   = blk / NC;
    const int c   = blk % NC;

    // Coordinate-space fold: fx = 0.5*((gx+1)*W - 1) = SX*gx + CX
    const float SX = 0.5f * (float)WW, CXc = 0.5f * (float)(WW - 1);
    const float SY = 0.5f * (float)HH, CYc = 0.5f * (float)(HH - 1);

    float acc[TPW];                          // per-lane channel-half partials
#pragma unroll
    for (int t = 0; t < TPW; ++t) acc[t] = 0.0f;

    const float* xb = x + (size_t)b * NIC * HWSZ;

    for (int chunk = 0; chunk < NCHUNK; ++chunk) {
        __syncthreads();                     // previous chunk fully consumed

        // Stage 16 planes x[b, chunk*16 .. +15, :] -> LDS (async, 49/thread).
        {
            const float* src = xb + (size_t)chunk * CPLANES * HWSZ;
#pragma unroll
            for (int k = 0; k < V4_PER_THR; ++k) {
                int idx = tid + k * WG;      // 0 .. 12543, exact cover
                async_ld_b128((unsigned)idx * 16u,
                              (const void*)((const char*)src + (size_t)idx * 16u));
            }
            WAIT_ASYNC(0);
        }
        __syncthreads();                     // all planes visible to all waves

        // --- Per-chunk WMMA A-operands (folded affine coefficients) --------
        // A is 16(M=i) x 4(K): lanes 0-15 hold {K0,K1} of row M=laneLo,
        // lanes 16-31 hold {K2,K3}. Both halves read theta of i = laneLo.
        const int iA = chunk * CPLANES + laneLo;
        const float* th = geo + ((size_t)c * NIC + (size_t)iA) * 6u;
        float t00 = th[0], t01 = th[1], t02 = th[2];
        float t10 = th[3], t11 = th[4], t12 = th[5];
        v2f Ax, Ay;
        Ax[0] = hiHalf ? (SX * t02 + CXc) : (SX * t00);
        Ax[1] = hiHalf ? 0.0f             : (SX * t01);
        Ay[0] = hiHalf ? (SY * t12 + CYc) : (SY * t10);
        Ay[1] = hiHalf ? 0.0f             : (SY * t11);

        // linComb for this lane's gather channels: i = chunk*16 + half*8 + r
        const float* lcp = lin + (size_t)c * NIC + (size_t)chunk * CPLANES
                               + (hiHalf ? 8 : 0);
        float lcv[8];
#pragma unroll
        for (int r = 0; r < 8; ++r) lcv[r] = lcp[r];

        const float* planeBase = smem + (hiHalf ? 8 * HWSZ : 0);

        // --- Tile loop: wave w owns tiles T = w, w+8, ... (exact 196) ------
#pragma unroll 1
        for (int t = 0; t < TPW; ++t) {
            const int T = wave + (t << 3);
            if (T >= NTILES) break;          // wave-uniform => EXEC stays full

            const int p  = T * 16 + laneLo;
            const int py = p / WW;
            const int px = p - py * WW;
            // Reciprocal-multiply (no IEEE fdiv chain in the hot loop).
            const float xsv = fmaf((float)px + 0.5f, INV_W2, -1.0f);
            const float ysv = fmaf((float)py + 0.5f, INV_H2, -1.0f);

            // B is 4(K) x 16(N=p): lanes 0-15 hold {K0,K1}=(xs,ys) of col p,
            // lanes 16-31 hold {K2,K3}=(1,0).
            v2f Bv;
            Bv[0] = hiHalf ? 1.0f : xsv;
            Bv[1] = hiHalf ? 0.0f : ysv;

            v8f cz = {};
            // fx(i,p), fy(i,p) for 16 channels x 16 pixels in two WMMAs.
            v8f Dx = __builtin_amdgcn_wmma_f32_16x16x4_f32(
                         false, Ax, false, Bv, (short)0, cz, false, false);
            v8f Dy = __builtin_amdgcn_wmma_f32_16x16x4_f32(
                         false, Ay, false, Bv, (short)0, cz, false, false);

            // Lane l<16: rows i=0..7; lane 16+l: rows i=8..15 (matches D layout).
            float accT = 0.0f;
#pragma unroll
            for (int r = 0; r < 8; ++r) {
                float s = bilin_gather(planeBase + r * HWSZ, Dx[r], Dy[r]);
                accT = fmaf(lcv[r], s, accT);
            }
            acc[t] += accT;
        }
    }

    // --- Epilogue: pairwise lane reduction, box term, coalesced store ------
    const float* bp = boxp + (size_t)c * 6u;
    const float bfx0 = SX * bp[0], bfx1 = SX * bp[1], bfx2 = SX * bp[2] + CXc;
    const float bfy0 = SY * bp[3], bfy1 = SY * bp[4], bfy2 = SY * bp[5] + CYc;
    float* outb = out + ((size_t)b * NC + (size_t)c) * HWSZ;

#pragma unroll 1
    for (int t = 0; t < TPW; ++t) {
        const int T = wave + (t << 3);
        if (T >= NTILES) break;
        float tot = acc[t] + __shfl_xor(acc[t], 16, 32);  // join channel halves
        if (!hiHalf) {
            const int p  = T * 16 + laneLo;
            const int py = p / WW;
            const int px = p - py * WW;
            const float xsv = fmaf((float)px + 0.5f, INV_W2, -1.0f);
            const float ysv = fmaf((float)py + 0.5f, INV_H2, -1.0f);
            float fx = bfx0 * xsv + bfx1 * ysv + bfx2;
            float fy = bfy0 * xsv + bfy1 * ysv + bfy2;
            outb[p] = tot * box_weight(fx, fy);
        }
    }
}

// ---------------------------------------------------------------------------
extern "C" void kernel_launch(void* const* d_in, const int* in_sizes, int n_in,
                              void* d_out, int out_size, void* d_ws, size_t ws_size,
                              hipStream_t stream) {
    (void)in_sizes; (void)n_in; (void)out_size; (void)d_ws; (void)ws_size;
    const float* x    = (const float*)d_in[0];
    const float* geo  = (const float*)d_in[1];
    const float* lin  = (const float*)d_in[2];
    const float* boxp = (const float*)d_in[3];
    float* out        = (float*)d_out;

    size_t shmem = (size_t)CPLANES * HWSZ * sizeof(float);   // 200704 B
    // CDNA5 allows up to 320 KB LDS per workgroup; raise the dynamic limit.
    (void)hipFuncSetAttribute((const void*)geom_layer_kernel,
                              hipFuncAttributeMaxDynamicSharedMemorySize,
                              (int)shmem);

    dim3 grid(NB * NC);   // 512 workgroups, b-major
    dim3 block(WG);
    hipLaunchKernelGGL(geom_layer_kernel, grid, block, shmem, stream,
                       x, geo, lin, boxp, out);
}